// SAGE_76725295775758
// MI455X (gfx1250) — compile-verified
//
#include <hip/hip_runtime.h>
#include <hip/hip_bf16.h>

typedef float v2f __attribute__((ext_vector_type(2)));
typedef float v8f __attribute__((ext_vector_type(8)));

#define DH 128

// ---------------------------------------------------------------------------
// Generic WMMA f32 GEMM: Y[r,o] = sum_k X[r,k] * W[o,k] + bias[o], optional ReLU.
// One wave per 16x16 output tile. K fixed = 128.
// f32 16x16x4 fragment layout (ISA 7.12.2):
//   A 16x4 : lanes 0-15 -> M=lane, {k,k+1}; lanes 16-31 -> M=lane-16, {k+2,k+3}
//   B 4x16 : lanes 0-15 -> N=lane, rows {k,k+1}; lanes 16-31 -> N=lane-16, rows {k+2,k+3}
//   C/D    : vgpr j -> M = j (lanes 0-15) / j+8 (lanes 16-31), N = lane&15
// ---------------------------------------------------------------------------
__global__ void gemm_wmma(const float* __restrict__ X, const float* __restrict__ W,
                          const float* __restrict__ bias, float* __restrict__ Y,
                          int nTiles, int OT, int doRelu) {
  int wave = threadIdx.x >> 5;
  int lane = threadIdx.x & 31;
  int tile = blockIdx.x * (blockDim.x >> 5) + wave;
  if (tile >= nTiles) return;                 // uniform per wave -> EXEC stays all-1s
  int row0 = (tile / OT) * 16;
  int o0   = (tile % OT) * 16;
  int m    = lane & 15;
  int koff = (lane >> 4) << 1;                // 0 or 2
  const float* xrow = X + (size_t)(row0 + m) * DH;
  const float* wrow = W + (size_t)(o0 + m) * DH;
  v8f c = {};
  #pragma unroll 4
  for (int kk = 0; kk < DH; kk += 4) {
    v2f a = *(const v2f*)(xrow + kk + koff);
    v2f b = *(const v2f*)(wrow + kk + koff);
    c = __builtin_amdgcn_wmma_f32_16x16x4_f32(false, a, false, b, (short)0, c, false, false);
  }
  int col   = o0 + m;
  int ldo   = OT * 16;
  float bo  = bias[col];
  int rbase = row0 + ((lane >> 4) << 3);
  #pragma unroll
  for (int j = 0; j < 8; ++j) {
    float v = c[j] + bo;
    if (doRelu) v = fmaxf(v, 0.0f);
    Y[(size_t)(rbase + j) * ldo + col] = v;
  }
}

// Fused SAGE layer GEMM: Y = agg*Wc^T + bc + cur*Ws^T + bs   (pre-norm output)
__global__ void conv_wmma(const float* __restrict__ agg, const float* __restrict__ cur,
                          const float* __restrict__ Wc, const float* __restrict__ bc,
                          const float* __restrict__ Ws, const float* __restrict__ bs,
                          float* __restrict__ Y, int nTiles) {
  const int OT = DH / 16;
  int wave = threadIdx.x >> 5;
  int lane = threadIdx.x & 31;
  int tile = blockIdx.x * (blockDim.x >> 5) + wave;
  if (tile >= nTiles) return;
  int row0 = (tile / OT) * 16;
  int o0   = (tile % OT) * 16;
  int m    = lane & 15;
  int koff = (lane >> 4) << 1;
  const float* arow  = agg + (size_t)(row0 + m) * DH;
  const float* xrow  = cur + (size_t)(row0 + m) * DH;
  const float* wcrow = Wc  + (size_t)(o0 + m) * DH;
  const float* wsrow = Ws  + (size_t)(o0 + m) * DH;
  v8f c = {};
  #pragma unroll 2
  for (int kk = 0; kk < DH; kk += 4) {
    v2f a0 = *(const v2f*)(arow  + kk + koff);
    v2f b0 = *(const v2f*)(wcrow + kk + koff);
    c = __builtin_amdgcn_wmma_f32_16x16x4_f32(false, a0, false, b0, (short)0, c, false, false);
    v2f a1 = *(const v2f*)(xrow  + kk + koff);
    v2f b1 = *(const v2f*)(wsrow + kk + koff);
    c = __builtin_amdgcn_wmma_f32_16x16x4_f32(false, a1, false, b1, (short)0, c, false, false);
  }
  int col   = o0 + m;
  float bo  = bc[col] + bs[col];
  int rbase = row0 + ((lane >> 4) << 3);
  #pragma unroll
  for (int j = 0; j < 8; ++j)
    Y[(size_t)(rbase + j) * DH + col] = c[j] + bo;
}

__global__ void zero_f32(float* __restrict__ p, size_t n) {
  size_t i = (size_t)blockIdx.x * blockDim.x + threadIdx.x;
  if (i < n) p[i] = 0.0f;
}

// One edge handled by 32 lanes; each lane atomically adds a float4 slice.
__global__ void edge_scatter(const float* __restrict__ x, const int* __restrict__ src,
                             const int* __restrict__ dst, float* __restrict__ agg,
                             float* __restrict__ deg, int E) {
  size_t t = (size_t)blockIdx.x * blockDim.x + threadIdx.x;
  int e = (int)(t >> 5);
  int lane = (int)(t & 31);
  if (e >= E) return;
  int s = src[e], d = dst[e];
  const float4 v = *(const float4*)(x + (size_t)s * DH + lane * 4);
  float* o = agg + (size_t)d * DH + lane * 4;
  atomicAdd(o + 0, v.x);
  atomicAdd(o + 1, v.y);
  atomicAdd(o + 2, v.z);
  atomicAdd(o + 3, v.w);
  if (lane == 0) atomicAdd(deg + d, 1.0f);
}

__global__ void inv_deg(float* __restrict__ deg, int n) {
  int i = blockIdx.x * blockDim.x + threadIdx.x;
  if (i < n) deg[i] = 1.0f / fmaxf(deg[i], 1.0f);
}

__global__ void scale_rows(float* __restrict__ agg, const float* __restrict__ invd, size_t total) {
  size_t i = (size_t)blockIdx.x * blockDim.x + threadIdx.x;
  if (i < total) agg[i] *= invd[i >> 7];
}

// Per-feature sum / sumsq over the node dimension (2 threads per feature per block).
__global__ void bn_stats(const float* __restrict__ pre, float* __restrict__ sum,
                         float* __restrict__ sq, int n) {
  int c = threadIdx.x & (DH - 1);
  int half = threadIdx.x >> 7;
  float s = 0.0f, q = 0.0f;
  for (int r = blockIdx.x * 2 + half; r < n; r += gridDim.x * 2) {
    float v = pre[(size_t)r * DH + c];
    s += v;
    q += v * v;
  }
  atomicAdd(&sum[c], s);
  atomicAdd(&sq[c], q);
}

__global__ void bn_finalize(const float* __restrict__ sum, const float* __restrict__ sq,
                            const float* __restrict__ gamma, const float* __restrict__ beta,
                            float* __restrict__ scale, float* __restrict__ shift, float invN) {
  int c = threadIdx.x;
  float mu  = sum[c] * invN;
  float var = sq[c] * invN - mu * mu;
  float sc  = gamma[c] * rsqrtf(var + 1e-5f);
  scale[c] = sc;
  shift[c] = beta[c] - mu * sc;
}

__global__ void bn_apply(const float* __restrict__ pre, const float* __restrict__ scale,
                         const float* __restrict__ shift, float* __restrict__ cur, size_t total) {
  size_t i = (size_t)blockIdx.x * blockDim.x + threadIdx.x;
  if (i >= total) return;
  int c = (int)(i & (DH - 1));
  cur[i] = fmaxf(pre[i] * scale[c] + shift[c], 0.0f);
}

extern "C" void kernel_launch(void* const* d_in, const int* in_sizes, int n_in,
                              void* d_out, int out_size, void* d_ws, size_t ws_size,
                              hipStream_t stream) {
  const float* x       = (const float*)d_in[0];
  const int*   ei      = (const int*)d_in[1];     // [2,E] int (x64-disabled JAX)
  const float* lin1_W  = (const float*)d_in[2];
  const float* lin1_b  = (const float*)d_in[3];
  const float* conv_W  = (const float*)d_in[4];   // [3,128,128]
  const float* conv_b  = (const float*)d_in[5];
  const float* skip_W  = (const float*)d_in[6];
  const float* skip_b  = (const float*)d_in[7];
  const float* bn_g    = (const float*)d_in[8];
  const float* bn_b    = (const float*)d_in[9];
  const float* lin2_W  = (const float*)d_in[10];  // [64,128]
  const float* lin2_b  = (const float*)d_in[11];

  const int N = in_sizes[0] / DH;                 // 50000
  const int E = in_sizes[1] / 2;                  // 800000
  const size_t ND = (size_t)N * DH;

  const int* src = ei;
  const int* dst = ei + E;

  // workspace layout
  float* cur   = (float*)d_ws;                    // [N,128] post-activation features
  float* agg   = cur + ND;                        // [N,128] mean-aggregation accumulator
  float* deg   = agg + ND;                        // [N]
  float* sum   = deg + N;                         // [128]
  float* sq    = sum + DH;                        // [128]
  float* scale = sq + DH;                         // [128]
  float* shift = scale + DH;                      // [128]

  float* h_out = (float*)d_out;                   // [N,128] pre-norm last-layer (also nxt scratch)
  float* y_out = h_out + ND;                      // [N,64]
  float* nxt   = h_out;                           // pre-norm buffer lives in d_out's h region

  const int rowTiles = N / 16;                    // 3125 (exact)
  const int WPB = 8;                              // waves per 256-thread block

  // 1) lin1 + relu  -> cur
  {
    int tiles = rowTiles * (DH / 16);
    gemm_wmma<<<(tiles + WPB - 1) / WPB, 256, 0, stream>>>(x, lin1_W, lin1_b, cur, tiles, DH / 16, 1);
  }

  const size_t zcount = ND + (size_t)N + 4 * DH;  // agg + deg + stats region
  for (int i = 0; i < 3; ++i) {
    zero_f32<<<(int)((zcount + 255) / 256), 256, 0, stream>>>(agg, zcount);

    size_t sthreads = (size_t)E * 32;
    edge_scatter<<<(int)((sthreads + 255) / 256), 256, 0, stream>>>(cur, src, dst, agg, deg, E);

    inv_deg<<<(N + 255) / 256, 256, 0, stream>>>(deg, N);
    scale_rows<<<(int)((ND + 255) / 256), 256, 0, stream>>>(agg, deg, ND);

    {
      int tiles = rowTiles * (DH / 16);
      conv_wmma<<<(tiles + WPB - 1) / WPB, 256, 0, stream>>>(
          agg, cur,
          conv_W + (size_t)i * DH * DH, conv_b + (size_t)i * DH,
          skip_W + (size_t)i * DH * DH, skip_b + (size_t)i * DH,
          nxt, tiles);
    }

    bn_stats<<<512, 256, 0, stream>>>(nxt, sum, sq, N);
    bn_finalize<<<1, DH, 0, stream>>>(sum, sq, bn_g + (size_t)i * DH, bn_b + (size_t)i * DH,
                                      scale, shift, 1.0f / (float)N);
    bn_apply<<<(int)((ND + 255) / 256), 256, 0, stream>>>(nxt, scale, shift, cur, ND);
    // after i==2, nxt (== d_out h region) already holds the pre-norm h output
  }

  // 2) lin2 -> y_out [N,64]
  {
    int OT = 64 / 16;
    int tiles = rowTiles * OT;
    gemm_wmma<<<(tiles + WPB - 1) / WPB, 256, 0, stream>>>(cur, lin2_W, lin2_b, y_out, tiles, OT, 0);
  }
}